// DependencyGNN_41094247088987
// MI455X (gfx1250) — compile-verified
//
#include <hip/hip_runtime.h>

typedef float v2f __attribute__((ext_vector_type(2)));
typedef float v8f __attribute__((ext_vector_type(8)));

#define N_NODES 50000
#define N_EDGES 600000
#define DIM     128
#define MTILES  (N_NODES / 16)   // 3125, exact

// ---------------- degree / norm precompute ----------------

__global__ void k_init_deg(float* __restrict__ deg) {
    int i = blockIdx.x * blockDim.x + threadIdx.x;
    if (i < N_NODES) deg[i] = 1.0f;            // self loop
}

__global__ void k_count(const int* __restrict__ dst, float* __restrict__ deg) {
    int e = blockIdx.x * blockDim.x + threadIdx.x;
    if (e < N_EDGES) atomicAdd(&deg[dst[e]], 1.0f);
}

__global__ void k_dinv(float* __restrict__ d) {
    int i = blockIdx.x * blockDim.x + threadIdx.x;
    if (i < N_NODES) d[i] = rsqrtf(d[i]);      // deg >= 1 always
}

// Repack W[k][n] (128x128 row-major) so an even/odd K pair of one column is a
// contiguous float2:  Wp[(k>>1)*256 + n*2 + (k&1)] = W[k][n]
__global__ void k_repack(const float* __restrict__ W, float* __restrict__ Wp) {
    int idx = blockIdx.x * blockDim.x + threadIdx.x;
    if (idx < DIM * DIM) {
        int k = idx >> 7, n = idx & 127;
        Wp[(k >> 1) * 256 + n * 2 + (k & 1)] = W[idx];
    }
}

// ---------------- fp32 WMMA GEMM: H[50000x128] = X[50000x128] @ W[128x128] ---
// One wave per 16-row M tile; 8 v8f accumulators cover all 128 output columns.
// Fragment layout (ISA 7.12.2, 16x16x4 f32):
//   A: lane l -> row (l&15), K = kb + 2*(l>>4) + {0,1}  (float2)
//   B: lane l -> col (l&15), same K pair                (float2, from Wp)
//   D: lane l, vgpr v -> row v + 8*(l>>4), col (l&15)

template <int RELU>
__global__ __launch_bounds__(256) void k_gemm(const float* __restrict__ X,
                                              const float* __restrict__ Wp,
                                              float* __restrict__ H) {
    int wave = threadIdx.x >> 5;
    int lane = threadIdx.x & 31;
    int mt   = blockIdx.x * 8 + wave;
    if (mt >= MTILES) return;

    int g = lane >> 4;      // lane group 0/1
    int r = lane & 15;      // row (A) / col (B,D) within tile

    const float* arow = X + ((size_t)mt * 16 + r) * DIM;

    v8f acc[8] = {};

#pragma unroll
    for (int kb = 0; kb < DIM; kb += 4) {
        int k0 = kb + 2 * g;                       // even -> 8B aligned
        v2f a = *(const v2f*)(arow + k0);
        if (RELU) { a.x = fmaxf(a.x, 0.0f); a.y = fmaxf(a.y, 0.0f); }
        const float* bbase = Wp + (size_t)(k0 >> 1) * 256 + r * 2;
#pragma unroll
        for (int nt = 0; nt < 8; ++nt) {
            v2f b = *(const v2f*)(bbase + nt * 32);
            acc[nt] = __builtin_amdgcn_wmma_f32_16x16x4_f32(
                false, a, false, b, (short)0, acc[nt], false, false);
        }
    }

    float* hrow = H + ((size_t)mt * 16 + 8 * g) * DIM + r;
#pragma unroll
    for (int v = 0; v < 8; ++v) {
#pragma unroll
        for (int nt = 0; nt < 8; ++nt)
            hrow[(size_t)v * DIM + nt * 16] = acc[nt][v];
    }
}

// out[i][c] = bias[c] + h[i][c] * dinv[i]^2   (self-loop term, no atomics)
__global__ void k_self(const float* __restrict__ h, const float* __restrict__ dinv,
                       const float* __restrict__ bias, float* __restrict__ out) {
    int idx = blockIdx.x * blockDim.x + threadIdx.x;
    if (idx < N_NODES * DIM) {
        int node = idx >> 7, c = idx & 127;
        float di = dinv[node];
        out[idx] = bias[c] + h[idx] * di * di;
    }
}

// one wave per edge: float4 gather of h[src], scaled atomic scatter to out[dst]
__global__ __launch_bounds__(256) void k_scatter(const float* __restrict__ h,
                                                 const int* __restrict__ src,
                                                 const int* __restrict__ dst,
                                                 const float* __restrict__ dinv,
                                                 float* __restrict__ out) {
    int tid  = blockIdx.x * blockDim.x + threadIdx.x;
    int e    = tid >> 5;
    int lane = threadIdx.x & 31;
    if (e >= N_EDGES) return;

    int s = src[e], d = dst[e];
    float nrm = dinv[s] * dinv[d];

    const float4* hv = (const float4*)(h + (size_t)s * DIM);
    float4 v = hv[lane];

    float* o = out + (size_t)d * DIM + lane * 4;
    atomicAdd(o + 0, v.x * nrm);
    atomicAdd(o + 1, v.y * nrm);
    atomicAdd(o + 2, v.z * nrm);
    atomicAdd(o + 3, v.w * nrm);
}

// ---------------- launch ----------------

extern "C" void kernel_launch(void* const* d_in, const int* in_sizes, int n_in,
                              void* d_out, int out_size, void* d_ws, size_t ws_size,
                              hipStream_t stream) {
    const float* X  = (const float*)d_in[0];
    const int*   ei = (const int*)d_in[1];     // [2, E]: row0 = src, row1 = dst
    const float* W1 = (const float*)d_in[2];
    const float* b1 = (const float*)d_in[3];
    const float* W2 = (const float*)d_in[4];
    const float* b2 = (const float*)d_in[5];
    float* out = (float*)d_out;

    const int* src = ei;
    const int* dst = ei + N_EDGES;

    float* ws   = (float*)d_ws;
    float* dinv = ws;                              // 50000 floats
    float* Wp1  = ws + 50048;                      // 16384 floats (aligned)
    float* Wp2  = Wp1 + 16384;                     // 16384 floats
    float* h    = Wp2 + 16384;                     // 6.4M floats
    float* x1   = h + (size_t)N_NODES * DIM;       // 6.4M floats

    const int thr = 256;
    const int gNode   = (N_NODES + thr - 1) / thr;        // 196
    const int gEdge1  = (N_EDGES + thr - 1) / thr;        // 2344 (1 edge/thread)
    const int gEdge32 = (N_EDGES * 32 + thr - 1) / thr;   // 75000 (1 edge/wave)
    const int gFeat   = (N_NODES * DIM + thr - 1) / thr;  // 25000
    const int gGemm   = (MTILES + 7) / 8;                 // 391
    const int gW      = (DIM * DIM + thr - 1) / thr;      // 64

    // normalization
    k_init_deg<<<gNode, thr, 0, stream>>>(dinv);
    k_count   <<<gEdge1, thr, 0, stream>>>(dst, dinv);
    k_dinv    <<<gNode, thr, 0, stream>>>(dinv);

    // weight repack for WMMA B fragments
    k_repack<<<gW, thr, 0, stream>>>(W1, Wp1);
    k_repack<<<gW, thr, 0, stream>>>(W2, Wp2);

    // layer 1: h = X @ W1 ; x1 = segsum(h*norm) + b1
    k_gemm<0><<<gGemm, thr, 0, stream>>>(X, Wp1, h);
    k_self    <<<gFeat, thr, 0, stream>>>(h, dinv, b1, x1);
    k_scatter <<<gEdge32, thr, 0, stream>>>(h, src, dst, dinv, x1);

    // layer 2: h = relu(x1) @ W2 ; out = segsum(h*norm) + b2
    k_gemm<1><<<gGemm, thr, 0, stream>>>(x1, Wp2, h);
    k_self    <<<gFeat, thr, 0, stream>>>(h, dinv, b2, out);
    k_scatter <<<gEdge32, thr, 0, stream>>>(h, src, dst, dinv, out);
}